// GCN_4827543241288
// MI455X (gfx1250) — compile-verified
//
#include <hip/hip_runtime.h>
#include <hip/hip_bf16.h>

typedef float v2f __attribute__((ext_vector_type(2)));
typedef float v8f __attribute__((ext_vector_type(8)));

#define N_FEAT_IN 128
#define HID 128
#define OUT_FEATS 64

// ---------------------------------------------------------------------------
// Utility: vectorized zero (n must be a multiple of 4 floats)
// ---------------------------------------------------------------------------
__global__ void gcn_zero4(float4* __restrict__ p, long long n4) {
    long long i = (long long)blockIdx.x * blockDim.x + threadIdx.x;
    if (i < n4) p[i] = make_float4(0.f, 0.f, 0.f, 0.f);
}

// ---------------------------------------------------------------------------
// Degree accumulation: deg_out[src[e]] += 1, deg_in[dst[e]] += 1
// ---------------------------------------------------------------------------
__global__ void gcn_degrees(const int* __restrict__ src, const int* __restrict__ dst,
                            float* __restrict__ deg_out, float* __restrict__ deg_in,
                            int nE) {
    int e = blockIdx.x * blockDim.x + threadIdx.x;
    if (e < nE) {
        atomicAdd(&deg_out[src[e]], 1.0f);
        atomicAdd(&deg_in[dst[e]], 1.0f);
    }
}

// In-place: deg -> clip(deg,1)^-0.5   (norm_src and norm_dst are contiguous)
__global__ void gcn_norms(float* __restrict__ deg, int n) {
    int i = blockIdx.x * blockDim.x + threadIdx.x;
    if (i < n) deg[i] = 1.0f / sqrtf(fmaxf(deg[i], 1.0f));
}

// ---------------------------------------------------------------------------
// SpMM scatter, 128 features: one wave (32 lanes x float4) per edge.
//   agg[dst] += x[src] * nsrc[src]
// Feature matrix (51.2 MB) is L2-resident (192 MB), so gathers + f32 atomics
// stay on-chip.
// ---------------------------------------------------------------------------
__global__ void gcn_spmm128(const int* __restrict__ src, const int* __restrict__ dst,
                            const float* __restrict__ x, const float* __restrict__ nsrc,
                            float* __restrict__ agg, int nE) {
    int wave = (int)(((long long)blockIdx.x * blockDim.x + threadIdx.x) >> 5);
    int lane = threadIdx.x & 31;
    if (wave >= nE) return;
    int s = src[wave];
    int d = dst[wave];
    float ns = nsrc[s];
    float4 v = ((const float4*)(x + (long long)s * N_FEAT_IN))[lane];
    float* ad = agg + (long long)d * N_FEAT_IN + lane * 4;
    atomicAdd(ad + 0, v.x * ns);
    atomicAdd(ad + 1, v.y * ns);
    atomicAdd(ad + 2, v.z * ns);
    atomicAdd(ad + 3, v.w * ns);
}

// SpMM scatter, 64 features (h already carries norm_src): 32 lanes x float2.
__global__ void gcn_spmm64(const int* __restrict__ src, const int* __restrict__ dst,
                           const float* __restrict__ x, float* __restrict__ agg, int nE) {
    int wave = (int)(((long long)blockIdx.x * blockDim.x + threadIdx.x) >> 5);
    int lane = threadIdx.x & 31;
    if (wave >= nE) return;
    int s = src[wave];
    int d = dst[wave];
    float2 v = ((const float2*)(x + (long long)s * OUT_FEATS))[lane];
    float* ad = agg + (long long)d * OUT_FEATS + lane * 2;
    atomicAdd(ad + 0, v.x);
    atomicAdd(ad + 1, v.y);
}

// ---------------------------------------------------------------------------
// Dense GEMM via V_WMMA_F32_16X16X4_F32 (full fp32 matrix pipe).
//   Out[M,NC] = epilogue( diag(pre?) * A[M,K] @ W[K,NC] )
// - Whole W staged in LDS once per block (W1=64KB, W2=32KB; WGP LDS=320KB).
// - One wave computes a 16x32 tile: two accumulators share one A fetch per
//   K-step (one global b64 load -> two v_wmma).
// - K compile-time -> fully unrolled 32-step chain of 64 WMMAs.
// A-operand layout (32-bit 16x4): lane l -> row (l&15), k = k0 + 2*(l>>4)+{0,1}
// B-operand layout (4x16):        lane l -> col (l&15), same k split.
// D layout: VGPR r -> M = r + 8*(lane>=16), N = lane&15.
// ---------------------------------------------------------------------------
template <int K, int NC, bool PRE, bool BIAS, bool POST, bool RELU>
__global__ __launch_bounds__(256) void gcn_gemm_wmma(
        const float* __restrict__ A, const float* __restrict__ W,
        float* __restrict__ Out,
        const float* __restrict__ pre_v,   // row scale applied to A
        const float* __restrict__ bias,    // per-column bias
        const float* __restrict__ post_v,  // row scale applied to result
        int n_rows) {
    __shared__ float sW[K * NC];

    // Stage W into LDS (coalesced float4, all 256 threads).
    const int tid = threadIdx.x;
#pragma unroll
    for (int i = tid; i < (K * NC) / 4; i += 256) {
        ((float4*)sW)[i] = ((const float4*)W)[i];
    }
    __syncthreads();

    const int lane = tid & 31;
    const int wave = tid >> 5;
    const int col_groups = NC / 32;
    const int tile_id = blockIdx.x * 8 + wave;
    const int total = (n_rows >> 4) * col_groups;

    if (tile_id < total) {  // wave-uniform guard: EXEC all-ones inside
        const int tm = tile_id / col_groups;
        const int cg = tile_id - tm * col_groups;
        const int row_base = tm << 4;
        const int col_base = cg << 5;

        const int l15 = lane & 15;
        const int khalf = (lane >> 4) << 1;  // 0 (lanes 0-15) or 2 (lanes 16-31)

        const float* arow = A + (long long)(row_base + l15) * K + khalf;
        const float pre = PRE ? pre_v[row_base + l15] : 1.0f;
        const float* wp = sW + khalf * NC + col_base + l15;

        v8f c0 = {};
        v8f c1 = {};
#pragma unroll
        for (int k = 0; k < K; k += 4) {
            v2f a, b0, b1;
            a.x = arow[0] * pre;
            a.y = arow[1] * pre;
            b0.x = wp[0];
            b0.y = wp[NC];
            b1.x = wp[16];
            b1.y = wp[NC + 16];
            c0 = __builtin_amdgcn_wmma_f32_16x16x4_f32(false, a, false, b0,
                                                       (short)0, c0, false, false);
            c1 = __builtin_amdgcn_wmma_f32_16x16x4_f32(false, a, false, b1,
                                                       (short)0, c1, false, false);
            arow += 4;
            wp += 4 * NC;
        }

        const float bv0 = BIAS ? bias[col_base + l15] : 0.0f;
        const float bv1 = BIAS ? bias[col_base + 16 + l15] : 0.0f;
        const int hi8 = (lane >> 4) << 3;  // 0 or 8
#pragma unroll
        for (int r = 0; r < 8; ++r) {
            int m = row_base + r + hi8;
            float v0 = c0[r] + bv0;
            float v1 = c1[r] + bv1;
            if (RELU) {
                v0 = fmaxf(v0, 0.0f);
                v1 = fmaxf(v1, 0.0f);
            }
            if (POST) {
                float p = post_v[m];
                v0 *= p;
                v1 *= p;
            }
            Out[(long long)m * NC + col_base + l15] = v0;
            Out[(long long)m * NC + col_base + 16 + l15] = v1;
        }
    }
}

// ---------------------------------------------------------------------------
// Final epilogue: out[i,j] = acc[i,j] * norm_dst[i] + b2[j]
// ---------------------------------------------------------------------------
__global__ void gcn_out_epilogue(const float* __restrict__ acc,
                                 const float* __restrict__ ndst,
                                 const float* __restrict__ b2,
                                 float* __restrict__ out, long long total) {
    long long idx = (long long)blockIdx.x * blockDim.x + threadIdx.x;
    if (idx < total) {
        int i = (int)(idx >> 6);  // OUT_FEATS == 64
        int j = (int)(idx & 63);
        out[idx] = acc[idx] * ndst[i] + b2[j];
    }
}

// ---------------------------------------------------------------------------
// Launch
// ---------------------------------------------------------------------------
extern "C" void kernel_launch(void* const* d_in, const int* in_sizes, int n_in,
                              void* d_out, int out_size, void* d_ws, size_t ws_size,
                              hipStream_t stream) {
    // Inputs (setup_inputs order): node_ids, src, dst, emb, W1, b1, W2, b2
    const int*   src = (const int*)d_in[1];
    const int*   dst = (const int*)d_in[2];
    const float* emb = (const float*)d_in[3];
    const float* W1  = (const float*)d_in[4];
    const float* b1  = (const float*)d_in[5];
    const float* W2  = (const float*)d_in[6];
    const float* b2  = (const float*)d_in[7];

    const int nN = in_sizes[0];
    const int nE = in_sizes[1];

    // Workspace layout (256B aligned):
    //   norm_src [N] | norm_dst [N] | B0 [N*128] | B1 [N*128]
    char* ws = (char*)d_ws;
    size_t off = 0;
    float* norm_src = (float*)(ws + off); off += ((size_t)nN * 4 + 255) & ~(size_t)255;
    float* norm_dst = (float*)(ws + off); off += ((size_t)nN * 4 + 255) & ~(size_t)255;
    float* B0       = (float*)(ws + off); off += ((size_t)nN * N_FEAT_IN * 4 + 255) & ~(size_t)255;
    float* B1       = (float*)(ws + off);

    float* out = (float*)d_out;

    const int T = 256;
    auto blocks4 = [](long long n4) { return (unsigned)((n4 + 255) / 256); };

    // 1) zero degree buffers (norm_src||norm_dst contiguous) and agg1 (B1)
    long long deg4 = ((long long)nN * 2) / 4;
    hipLaunchKernelGGL(gcn_zero4, dim3(blocks4(deg4)), dim3(T), 0, stream,
                       (float4*)norm_src, deg4);
    long long agg4 = (long long)nN * N_FEAT_IN / 4;
    hipLaunchKernelGGL(gcn_zero4, dim3(blocks4(agg4)), dim3(T), 0, stream,
                       (float4*)B1, agg4);

    // 2) degrees + norms
    hipLaunchKernelGGL(gcn_degrees, dim3((nE + T - 1) / T), dim3(T), 0, stream,
                       src, dst, norm_src, norm_dst, nE);
    hipLaunchKernelGGL(gcn_norms, dim3((2 * nN + T - 1) / T), dim3(T), 0, stream,
                       norm_src, 2 * nN);

    // 3) SpMM1: B1[dst] += emb[src] * norm_src[src]   (one wave per edge)
    long long spmm1_threads = (long long)nE * 32;
    hipLaunchKernelGGL(gcn_spmm128, dim3((unsigned)((spmm1_threads + T - 1) / T)), dim3(T),
                       0, stream, src, dst, emb, norm_src, B1, nE);

    // 4) GEMM1 (WMMA fp32): h1 = relu( (B1 * norm_dst[row]) @ W1 + b1 ) -> B0
    {
        int total_tiles = (nN >> 4) * (HID / 32);  // 6250 * 4 wave-tiles (16x32 each)
        hipLaunchKernelGGL((gcn_gemm_wmma<N_FEAT_IN, HID, true, true, false, true>),
                           dim3((total_tiles + 7) / 8), dim3(T), 0, stream,
                           B1, W1, B0, norm_dst, b1, (const float*)nullptr, nN);
    }

    // 5) GEMM2 (WMMA fp32): h2 = (B0 @ W2) * norm_src[row] -> B1 [N*64]
    {
        int total_tiles = (nN >> 4) * (OUT_FEATS / 32);  // 6250 * 2
        hipLaunchKernelGGL((gcn_gemm_wmma<HID, OUT_FEATS, false, false, true, false>),
                           dim3((total_tiles + 7) / 8), dim3(T), 0, stream,
                           B0, W2, B1, (const float*)nullptr, (const float*)nullptr,
                           norm_src, nN);
    }

    // 6) zero acc2 (reuse B0, first N*64 floats), then SpMM2: B0[dst] += B1[src]
    long long acc4 = (long long)nN * OUT_FEATS / 4;
    hipLaunchKernelGGL(gcn_zero4, dim3(blocks4(acc4)), dim3(T), 0, stream,
                       (float4*)B0, acc4);
    long long spmm2_threads = (long long)nE * 32;
    hipLaunchKernelGGL(gcn_spmm64, dim3((unsigned)((spmm2_threads + T - 1) / T)), dim3(T),
                       0, stream, src, dst, B1, B0, nE);

    // 7) out = B0 * norm_dst[row] + b2
    long long total = (long long)nN * OUT_FEATS;
    hipLaunchKernelGGL(gcn_out_epilogue, dim3((unsigned)((total + T - 1) / T)), dim3(T),
                       0, stream, B0, norm_dst, b2, out, total);
}